// PairBilinearHalve_83365315216074
// MI455X (gfx1250) — compile-verified
//
#include <hip/hip_runtime.h>

typedef __attribute__((ext_vector_type(2))) float v2f;
typedef __attribute__((ext_vector_type(4))) float v4f;
typedef __attribute__((ext_vector_type(4))) int   v4i;

// AS-qualified int4 pointers for the async global->LDS builtin
typedef __attribute__((address_space(1))) v4i* as1_b128_t;
typedef __attribute__((address_space(3))) v4i* as3_b128_t;

#define G       64          // grid side
#define NPAIR   2048
#define BATCH   8192
#define DIMX    4096        // 2*NPAIR
#define PBLK    16          // pairs staged per workgroup
#define SLAB    4100        // padded floats per pair slab (64*64 + 4 -> bank rotate by 4)
#define THREADS 512         // 16 wave32
#define BROWS   (THREADS / PBLK)   // 32 batch rows in flight
#define BCHUNK  1024        // batch rows per workgroup
#define STAGE_ITERS ((PBLK * (G * G / 4)) / THREADS)   // 16384 float4 / 512 thr = 32

#ifndef __has_builtin
#define __has_builtin(x) 0
#endif

#if __has_builtin(__builtin_amdgcn_global_load_async_to_lds_b128)
#define ASYNC_LDS 1
#else
#define ASYNC_LDS 0
#endif

__global__ __launch_bounds__(THREADS)
void PairBilinear_kernel(const float* __restrict__ x,
                         const float* __restrict__ W,
                         const float* __restrict__ Y,
                         float* __restrict__ out)
{
    extern __shared__ float lds[];   // PBLK * SLAB floats = 262400 B (fits 320KB WGP LDS)

    const int tid    = threadIdx.x;
    const int p0     = blockIdx.x * PBLK;
    const int bstart = blockIdx.y * BCHUNK;

    // ---- Stage PBLK pair tables (64x64 f32 each) into LDS, 16B chunks ----
    // Constant trip count, fully unrolled: straight-line async issues, no
    // exec-mask loop machinery around the EXEC-sensitive async ops.
    const float* ysrc = Y + (size_t)p0 * (G * G);
    #pragma unroll
    for (int i = 0; i < STAGE_ITERS; ++i) {
        const int idx  = i * THREADS + tid;
        const int slab = idx >> 10;              // (G*G/4) = 1024 float4 per slab
        const int offf = (idx & 1023) << 2;      // float offset within slab
#if ASYNC_LDS
        __builtin_amdgcn_global_load_async_to_lds_b128(
            (as1_b128_t)(ysrc + slab * (G * G) + offf),
            (as3_b128_t)&lds[slab * SLAB + offf],
            0, 0);
#else
        *(v4f*)&lds[slab * SLAB + offf] =
            *(const v4f*)(ysrc + slab * (G * G) + offf);
#endif
    }
#if ASYNC_LDS
#if __has_builtin(__builtin_amdgcn_s_wait_asynccnt)
    __builtin_amdgcn_s_wait_asynccnt(0);
#else
    asm volatile("s_wait_asynccnt 0" ::: "memory");
#endif
#endif
    __syncthreads();

    // ---- Per-thread: fixed pair, march over batch rows ----
    const int pp   = tid & (PBLK - 1);
    const int brow = tid >> 4;                   // log2(PBLK)
    const int p    = p0 + pp;

    const float w00 = W[p * 4 + 0];              // W[p][0][0]
    const float w01 = W[p * 4 + 1];              // W[p][0][1]
    const float w10 = W[p * 4 + 2];              // W[p][1][0]
    const float w11 = W[p * 4 + 3];              // W[p][1][1]
    const float* lY = lds + pp * SLAB;

    #pragma unroll 4
    for (int it = 0; it < BCHUNK / BROWS; ++it) {
        const int b = bstart + it * BROWS + brow;

        // coalesced 8B/lane streaming load (non-temporal: keep L2 for Y)
        v2f xv = __builtin_nontemporal_load(
            (const v2f*)(x + (size_t)b * DIMX + p * 2));

        // 2x2 matvec: out_j = sum_i x_i * W[i][j]
        const float xr = xv.x * w00 + xv.y * w10;   // j=0 -> row coord
        const float xc = xv.x * w01 + xv.y * w11;   // j=1 -> col coord

        float gr = fminf(fmaxf(xr * (float)(G - 1), 0.0f), (float)(G - 1));
        float gc = fminf(fmaxf(xc * (float)(G - 1), 0.0f), (float)(G - 1));
        int r0 = (int)gr; r0 = (r0 > G - 2) ? (G - 2) : r0;   // gr >= 0 so trunc == floor
        int c0 = (int)gc; c0 = (c0 > G - 2) ? (G - 2) : c0;
        const float fr = gr - (float)r0;
        const float fc = gc - (float)c0;

        // LDS gather (bank-rotated slabs)
        const float* cell = lY + r0 * G + c0;
        const float y00 = cell[0];
        const float y01 = cell[1];
        const float y10 = cell[G];
        const float y11 = cell[G + 1];

        const float res = y00 * (1.0f - fr) * (1.0f - fc)
                        + y01 * (1.0f - fr) * fc
                        + y10 * fr * (1.0f - fc)
                        + y11 * fr * fc;

        __builtin_nontemporal_store(res, out + (size_t)b * NPAIR + p);
    }
}

extern "C" void kernel_launch(void* const* d_in, const int* in_sizes, int n_in,
                              void* d_out, int out_size, void* d_ws, size_t ws_size,
                              hipStream_t stream) {
    const float* x = (const float*)d_in[0];   // (8192, 4096)
    const float* W = (const float*)d_in[1];   // (2048, 2, 2)
    const float* Y = (const float*)d_in[2];   // (2048, 64, 64)
    float* out = (float*)d_out;               // (8192, 2048)

    dim3 grid(NPAIR / PBLK, BATCH / BCHUNK);  // 128 x 8
    size_t shmem = (size_t)PBLK * SLAB * sizeof(float);  // 262400 B
    hipLaunchKernelGGL(PairBilinear_kernel, grid, dim3(THREADS), shmem, stream,
                       x, W, Y, out);
}